// LiangCausalityEstimator_87720412054227
// MI455X (gfx1250) — compile-verified
//
#include <hip/hip_runtime.h>
#include <hip/hip_bf16.h>
#include <stdint.h>

#define BATCH 64
#define NX    64
#define NT    8192
#define MCOL  8191          // columns of Xp / dX
#define DENOM 8190.0f
#define KC    128           // k-chunk staged in LDS
#define LROW  132           // staged floats per row: 128 core + 1 tail + pad (16B-aligned rows)

typedef __attribute__((ext_vector_type(2))) float v2f;
typedef __attribute__((ext_vector_type(8))) float v8f;

// workspace layout per batch (floats)
#define OFF_G    0
#define OFF_H    4096
#define OFF_C    8192
#define OFF_RXP  12288
#define OFF_SX   16384
#define OFF_SD   16448
#define OFF_D2   16512
#define OFF_INVD 16576
#define OFF_IB2  16640
#define OFF_IB3  16704
#define OFF_RS1  16768
#define OFF_QQ   16832
#define WSB      16896

// ---------------------------------------------------------------------------
// CDNA5 async global->LDS copies (ASYNCcnt-tracked, VGPR-bypass).
// LDS operand = workgroup-relative byte offset (low 32 bits of generic ptr).
// ---------------------------------------------------------------------------
__device__ __forceinline__ void async_g2l_b128(const void* gptr, void* lptr) {
  uint32_t loff = (uint32_t)(uintptr_t)lptr;
  asm volatile("global_load_async_to_lds_b128 %0, %1, off"
               :: "v"(loff), "v"(gptr) : "memory");
}
__device__ __forceinline__ void async_g2l_b32(const void* gptr, void* lptr) {
  uint32_t loff = (uint32_t)(uintptr_t)lptr;
  asm volatile("global_load_async_to_lds_b32 %0, %1, off"
               :: "v"(loff), "v"(gptr) : "memory");
}
__device__ __forceinline__ void wait_async0() {
  asm volatile("s_wait_asynccnt 0x0" ::: "memory");
}

// ---------------------------------------------------------------------------
// Stage 1: raw Grams  G = Xp Xp^T,  H = Xp dX^T  via V_WMMA_F32_16X16X4_F32.
// One workgroup (8 wave32) per batch; each wave owns 4 16x16 output tiles.
// ---------------------------------------------------------------------------
__global__ __launch_bounds__(256) void k_gram(const float* __restrict__ X,
                                              float* __restrict__ ws) {
  __shared__ float Xs[NX * LROW];          // 64 x 132 floats = 33.8 KB
  const int b    = blockIdx.x;
  const int tid  = threadIdx.x;
  const int lane = tid & 31;
  const int w    = tid >> 5;               // wave id 0..7 (wave32)
  const float* Xb = X + (size_t)b * NX * NT;
  float* wsb = ws + (size_t)b * WSB;

  const int rBlk = (w >> 1) * 16;          // A row block (rows of Xp)
  const int cBlk = (w & 1) * 32;           // B col blocks cBlk, cBlk+16

  v8f g0 = {}, g1 = {}, h0 = {}, h1 = {};

  const int nChunks = (MCOL + KC - 1) / KC;   // 64
  const int mOff2 = (lane >> 4) << 1;         // lanes 0-15 -> K{0,1}, 16-31 -> K{2,3}
  const int rA  = rBlk + (lane & 15);
  const int rB0 = cBlk + (lane & 15);
  const int rB1 = cBlk + 16 + (lane & 15);

  for (int ch = 0; ch < nChunks; ++ch) {
    const int kb = ch * KC;

    // --- async-stage X[:, kb .. kb+KC] into LDS ---
    // core: 64 rows x 128 floats as aligned B128 async copies (always in-range)
    for (int idx = tid; idx < NX * (KC / 4); idx += 256) {
      int r = idx >> 5, c4 = (idx & 31) << 2;
      async_g2l_b128(Xb + (size_t)r * NT + kb + c4, &Xs[r * LROW + c4]);
    }
    // tail column kb+KC (needed for the last dX of the chunk)
    if (tid < NX) {
      int gc = kb + KC;
      if (gc < NT) async_g2l_b32(Xb + (size_t)tid * NT + gc, &Xs[tid * LROW + KC]);
      else         Xs[tid * LROW + KC] = 0.0f;   // DScnt store, drained by barrier
    }
    wait_async0();
    __syncthreads();

    // prefetch next chunk (global_prefetch_b8)
    if (ch + 1 < nChunks) {
      int pr = tid >> 2;
      int gc = (ch + 1) * KC + ((tid & 3) << 5);
      if (gc < NT) __builtin_prefetch(&Xb[(size_t)pr * NT + gc], 0, 1);
    }

    for (int kl = 0; kl < KC; kl += 4) {
      const int kk = kl + mOff2;
      const int gk = kb + kk;
      const bool v0 = gk < MCOL;
      const bool v1 = (gk + 1) < MCOL;

      // A fragment: Xp rows rBlk..rBlk+15 (16x4 fp32 layout)
      const float* pa = &Xs[rA * LROW + kk];
      float a0 = pa[0], a1 = pa[1];
      v2f fa; fa.x = v0 ? a0 : 0.0f; fa.y = v1 ? a1 : 0.0f;

      // B fragments (4x16 layout): Xp and dX for col block 0
      const float* p0 = &Xs[rB0 * LROW + kk];
      float b0 = p0[0], b1 = p0[1], b2 = p0[2];
      v2f fb0, fd0;
      fb0.x = v0 ? b0 : 0.0f;        fb0.y = v1 ? b1 : 0.0f;
      fd0.x = v0 ? (b1 - b0) : 0.0f; fd0.y = v1 ? (b2 - b1) : 0.0f;

      // col block 1
      const float* p1 = &Xs[rB1 * LROW + kk];
      float c0 = p1[0], c1 = p1[1], c2 = p1[2];
      v2f fb1, fd1;
      fb1.x = v0 ? c0 : 0.0f;        fb1.y = v1 ? c1 : 0.0f;
      fd1.x = v0 ? (c1 - c0) : 0.0f; fd1.y = v1 ? (c2 - c1) : 0.0f;

      g0 = __builtin_amdgcn_wmma_f32_16x16x4_f32(false, fa, false, fb0, (short)0, g0, false, false);
      g1 = __builtin_amdgcn_wmma_f32_16x16x4_f32(false, fa, false, fb1, (short)0, g1, false, false);
      h0 = __builtin_amdgcn_wmma_f32_16x16x4_f32(false, fa, false, fd0, (short)0, h0, false, false);
      h1 = __builtin_amdgcn_wmma_f32_16x16x4_f32(false, fa, false, fd1, (short)0, h1, false, false);
    }
    __syncthreads();
  }

  // C/D layout: VGPR v, lanes 0-15 -> M=v, lanes 16-31 -> M=v+8; N = lane&15
  const int mOff = (lane >= 16) ? 8 : 0;
  const int n = lane & 15;
#pragma unroll
  for (int v = 0; v < 8; ++v) {
    int row = rBlk + v + mOff;
    wsb[OFF_G + row * 64 + cBlk + n]      = g0[v];
    wsb[OFF_G + row * 64 + cBlk + 16 + n] = g1[v];
    wsb[OFF_H + row * 64 + cBlk + n]      = h0[v];
    wsb[OFF_H + row * 64 + cBlk + 16 + n] = h1[v];
  }
}

// ---------------------------------------------------------------------------
// Stage 1b: per-row reductions: sx = sum Xp, D2 = sum dX^2, sd = X[-1]-X[0]
// ---------------------------------------------------------------------------
__global__ __launch_bounds__(256) void k_rowsum(const float* __restrict__ X,
                                                float* __restrict__ ws) {
  __shared__ float red[256];
  __shared__ float red2[256];
  const int b = blockIdx.x >> 6;
  const int i = blockIdx.x & 63;
  const float* row = X + ((size_t)b * NX + i) * NT;
  float s = 0.0f, q = 0.0f;
  for (int k = threadIdx.x; k < MCOL; k += 256) {
    float x0 = row[k], x1 = row[k + 1];
    s += x0;
    float d = x1 - x0;
    q += d * d;
  }
  red[threadIdx.x] = s;
  red2[threadIdx.x] = q;
  __syncthreads();
  for (int off = 128; off > 0; off >>= 1) {
    if (threadIdx.x < off) {
      red[threadIdx.x]  += red[threadIdx.x + off];
      red2[threadIdx.x] += red2[threadIdx.x + off];
    }
    __syncthreads();
  }
  if (threadIdx.x == 0) {
    float* wsb = ws + (size_t)b * WSB;
    wsb[OFF_SX + i] = red[0];
    wsb[OFF_D2 + i] = red2[0];
    wsb[OFF_SD + i] = row[NT - 1] - row[0];
  }
}

// ---------------------------------------------------------------------------
// Stage 2: per batch — build C, dC; Gauss-Jordan solve C P = dC; epilogues.
// ---------------------------------------------------------------------------
__global__ __launch_bounds__(256) void k_solve(float* __restrict__ ws,
                                               float* __restrict__ out) {
  __shared__ float Aug[64][130];   // [C | dC] -> [I | P]
  __shared__ float Cs[64][65];     // pristine copy of C
  __shared__ float fac[64];
  __shared__ float sxs[64], sds[64], invd[64], dHn[64], ZZ[64];
  const int tid = threadIdx.x;
  const int b = blockIdx.x;
  float* wsb = ws + (size_t)b * WSB;
  const float Mf = (float)MCOL;

  if (tid < 64) { sxs[tid] = wsb[OFF_SX + tid]; sds[tid] = wsb[OFF_SD + tid]; }
  __syncthreads();

  for (int idx = tid; idx < 64 * 128; idx += 256) {
    int i = idx >> 7, c = idx & 127;
    if (c < 64) {
      float v = (wsb[OFF_G + i * 64 + c] - sxs[i] * sxs[c] / Mf) / DENOM;
      Aug[i][c] = v;
      Cs[i][c] = v;
    } else {
      int j = c - 64;
      Aug[i][c] = (wsb[OFF_H + i * 64 + j] - sxs[i] * sds[j] / Mf) / DENOM;
    }
  }
  __syncthreads();

  // Gauss-Jordan (no pivoting; C is SPD covariance)
  for (int k = 0; k < 64; ++k) {
    float ipiv = 1.0f / Aug[k][k];
    if (tid < 64) fac[tid] = Aug[tid][k];
    __syncthreads();
    for (int c = tid; c < 128; c += 256) Aug[k][c] *= ipiv;
    __syncthreads();
    for (int idx = tid; idx < 64 * 128; idx += 256) {
      int r = idx >> 7, c = idx & 127;
      if (r != k) Aug[r][c] -= fac[r] * Aug[k][c];
    }
    __syncthreads();
  }
  // now P[i][j] = Aug[i][64+j]

  if (tid < 64) invd[tid] = 1.0f / Cs[tid][tid];
  if (tid < 64) {
    const int i = tid;
    float t1 = 0.0f;
    for (int j = 0; j < 64; ++j) t1 += sxs[j] * Aug[j][64 + i];
    const float ff = (sds[i] - t1) / Mf;

    float pgp = 0.0f, htp = 0.0f;
    for (int j = 0; j < 64; ++j) {
      float u = 0.0f;
      for (int l = 0; l < 64; ++l) u += wsb[OFF_G + l * 64 + j] * Aug[l][64 + i];
      pgp += Aug[j][64 + i] * u;
      htp += wsb[OFF_H + j * 64 + i] * Aug[j][64 + i];
      // RR @ Xp^T row i  =  H^T - ff*sx - P^T G
      wsb[OFF_RXP + i * 64 + j] = wsb[OFF_H + j * 64 + i] - ff * sxs[j] - u;
    }
    const float qq  = wsb[OFF_D2 + i] + Mf * ff * ff + pgp
                    - 2.0f * ff * sds[i] - 2.0f * htp + 2.0f * ff * t1;
    const float rs1 = sds[i] - Mf * ff - t1;
    const float bb2 = qq / Mf;           // QQ*DT/(nt-N_STEP)
    const float bb  = sqrtf(bb2);
    const float ib2 = 1.0f / bb2;
    const float ib3 = ib2 / bb;
    wsb[OFF_QQ + i]   = qq;
    wsb[OFF_RS1 + i]  = rs1;
    wsb[OFF_IB2 + i]  = ib2;
    wsb[OFF_IB3 + i]  = ib3;
    wsb[OFF_INVD + i] = invd[i];
    dHn[i] = bb2 * 0.5f * invd[i];       // bb^2 / (2*diagC)
  }
  __syncthreads();

  if (tid < 64) {
    const int j = tid;
    float z = fabsf(dHn[j]);
    const float idj = invd[j];
    for (int i2 = 0; i2 < 64; ++i2)
      z += fabsf(Cs[i2][j] * idj * Aug[i2][64 + j]);
    ZZ[j] = z;
  }
  __syncthreads();

  for (int idx = tid; idx < 4096; idx += 256) {
    int i = idx >> 6, j = idx & 63;
    float cm = Cs[i][j] * invd[j] * Aug[i][64 + j];
    out[(size_t)b * 4096 + idx] = cm;                                      // cM
    out[(size_t)(2 * BATCH) * 4096 + (size_t)b * 4096 + idx] = cm / ZZ[j]; // cM_Z
    wsb[OFF_C + idx] = Cs[i][j];
  }
}

// ---------------------------------------------------------------------------
// Stage 3: one workgroup per (batch, i): build 66x66 NNI, Gauss-Jordan
// inverse in LDS, take inner diagonal, emit var.T
// ---------------------------------------------------------------------------
#define NN   66
#define NAUG 132
__global__ __launch_bounds__(256) void k_invert(const float* __restrict__ ws,
                                                float* __restrict__ out) {
  __shared__ float A[NN][NAUG + 1];
  __shared__ float fac[NN];
  const int tid = threadIdx.x;
  const int b = blockIdx.x >> 6;
  const int i = blockIdx.x & 63;
  const float* wsb = ws + (size_t)b * WSB;
  const float ib2 = wsb[OFF_IB2 + i];
  const float ib3 = wsb[OFF_IB3 + i];
  const float rs1 = wsb[OFF_RS1 + i];
  const float qq  = wsb[OFF_QQ + i];
  const float Nf  = 8191.0f;

  for (int idx = tid; idx < NN * NAUG; idx += 256) {
    int r = idx / NAUG, c = idx - r * NAUG;
    float v;
    if (c >= NN) {
      v = ((c - NN) == r) ? 1.0f : 0.0f;
    } else if (r >= 1 && r <= 64 && c >= 1 && c <= 64) {
      v = ib2 * wsb[OFF_G + (r - 1) * 64 + (c - 1)];
    } else if (r == 0 && c >= 1 && c <= 64) {
      v = ib2 * wsb[OFF_SX + (c - 1)];
    } else if (c == 0 && r >= 1 && r <= 64) {
      v = ib2 * wsb[OFF_SX + (r - 1)];
    } else if (c == 65 && r >= 1 && r <= 64) {
      v = 2.0f * ib3 * wsb[OFF_RXP + i * 64 + (r - 1)];
    } else if (r == 65 && c >= 1 && c <= 64) {
      v = 2.0f * ib3 * wsb[OFF_RXP + i * 64 + (c - 1)];
    } else if (r == 0 && c == 0) {
      v = Nf * ib2;
    } else if ((r == 0 && c == 65) || (r == 65 && c == 0)) {
      v = 2.0f * ib3 * rs1;
    } else {  // r==65 && c==65
      v = 3.0f * qq * ib2 * ib2 - Nf * ib2;
    }
    A[r][c] = v;
  }
  __syncthreads();

  for (int k = 0; k < NN; ++k) {
    float ipiv = 1.0f / A[k][k];
    if (tid < NN) fac[tid] = A[tid][k];
    __syncthreads();
    for (int c = tid; c < NAUG; c += 256) A[k][c] *= ipiv;
    __syncthreads();
    for (int idx = tid; idx < NN * NAUG; idx += 256) {
      int r = idx / NAUG, c = idx - r * NAUG;
      if (r != k) A[r][c] -= fac[r] * A[k][c];
    }
    __syncthreads();
  }

  if (tid < 64) {
    const int j = tid;
    float dinner = A[1 + j][NN + 1 + j];
    float t = wsb[OFF_INVD + i] * wsb[OFF_C + i * 64 + j];
    // var.T -> out[BATCH*4096 + b*4096 + j*64 + i]
    out[(size_t)BATCH * 4096 + (size_t)b * 4096 + j * 64 + i] = t * t * dinner;
  }
}

// ---------------------------------------------------------------------------
extern "C" void kernel_launch(void* const* d_in, const int* in_sizes, int n_in,
                              void* d_out, int out_size, void* d_ws, size_t ws_size,
                              hipStream_t stream) {
  (void)in_sizes; (void)n_in; (void)out_size; (void)ws_size;
  const float* X = (const float*)d_in[0];
  float* out = (float*)d_out;
  float* ws  = (float*)d_ws;

  k_gram  <<<dim3(BATCH),      dim3(256), 0, stream>>>(X, ws);
  k_rowsum<<<dim3(BATCH * 64), dim3(256), 0, stream>>>(X, ws);
  k_solve <<<dim3(BATCH),      dim3(256), 0, stream>>>(ws, out);
  k_invert<<<dim3(BATCH * 64), dim3(256), 0, stream>>>(ws, out);
}